// AdvancedFrequencyDomainIPNLMS_78357383349001
// MI455X (gfx1250) — compile-verified
//
#include <hip/hip_runtime.h>
#include <hip/hip_bf16.h>
#include <stdint.h>

// ---------------------------------------------------------------------------
// Frequency-domain dual-filter IPNLMS AEC for MI455X (gfx1250, wave32).
//
// Mapping: one workgroup, 160 threads (5 wave32s), one frequency bin per lane
// (lanes 129..159 mirror bin 128 and never store). The 16000-frame recurrence
// runs in registers; the only memory dependency per step (4 floats/bin) is
// hidden with an 8-deep LDS prefetch ring fed by global_load_async_to_lds_b32
// (ASYNCcnt) and consumed behind s_wait_asynccnt 28.
// ---------------------------------------------------------------------------

#define F_BINS   129
#define K_TAPS   10
#define NBATCH   8
#define NT       2000
#define NSTEPS   (NBATCH * NT)     // 16000 serial frames
#define THREADS  160               // 5 wave32s
#define DEPTH    8                 // prefetch ring slots (4 async loads each)
#define MIDP1    36                // AEC_MID_CHAN + 1 -> nblocks = 10 below, 8 above

typedef float v2f __attribute__((ext_vector_type(2)));   // native vector for NT store

// CDNA5 async global->LDS copy (tracked by ASYNCcnt). GV addressing mode:
//   global_load_async_to_lds_b32 vdst(LDS byte addr), vaddr64, off
__device__ __forceinline__ void async_ld_b32(uint32_t lds_byte, const float* g) {
  asm volatile("global_load_async_to_lds_b32 %0, %1, off"
               :: "v"(lds_byte), "v"((uint64_t)(uintptr_t)g)
               : "memory");
}

// Oldest ring slot (4 loads) complete once <= 4*(DEPTH-1)=28 remain outstanding
// (async loads complete in order within a wave).
__device__ __forceinline__ void wait_async_le28() {
  asm volatile("s_wait_asynccnt 28" ::: "memory");
}
// Order the ds_load of a slot before its async overwrite is issued.
__device__ __forceinline__ void wait_ds0() {
  asm volatile("s_wait_dscnt 0" ::: "memory");
}

__device__ __forceinline__ void issue_prefetch(uint32_t ldsLane, int p, size_t fc,
                                               const float* __restrict__ micR,
                                               const float* __restrict__ micI,
                                               const float* __restrict__ refR,
                                               const float* __restrict__ refI) {
  // frame p: t = p/8, b = p%8; element (b*NT + t)*F + f of each [B,T,F] array
  const int t = p >> 3, b = p & 7;
  const size_t base = (size_t)(b * NT + t) * F_BINS + fc;
  const uint32_t la = ldsLane + (uint32_t)(p & (DEPTH - 1)) * (THREADS * 16);
  async_ld_b32(la + 0,  micR + base);
  async_ld_b32(la + 4,  micI + base);
  async_ld_b32(la + 8,  refR + base);
  async_ld_b32(la + 12, refI + base);
}

__global__ __launch_bounds__(THREADS, 1)
void aec_ipnlms_kernel(const float* __restrict__ micR, const float* __restrict__ micI,
                       const float* __restrict__ refR, const float* __restrict__ refI,
                       const float* __restrict__ fr0,  const float* __restrict__ fi0,
                       const float* __restrict__ ar0,  const float* __restrict__ ai0,
                       float* __restrict__ out)
{
  __shared__ float4 frameBuf[DEPTH * THREADS];   // [slot][lane] = {mr, mi, rr, ri}

  const int   tid   = threadIdx.x;
  const int   fc    = (tid < F_BINS) ? tid : (F_BINS - 1);
  const bool  nb10  = (fc < MIDP1);              // 10 taps below, 8 above
  const float inv2nb = nb10 ? (0.25f / 10.0f) : (0.25f / 8.0f); // (1-a)/(2*nb), a=0.5

  // Low 32 bits of the generic address of a __shared__ object == LDS byte offset.
  const uint32_t ldsLane = (uint32_t)(uintptr_t)(&frameBuf[tid]);

  // ---- per-lane recurrence state, all in registers --------------------------
  float hr[K_TAPS], hi[K_TAPS];                  // reference history
  float fr[K_TAPS], fi[K_TAPS];                  // main (FIR) filter
  float ar[K_TAPS], ai[K_TAPS];                  // adaptive (ADF) filter
#pragma unroll
  for (int k = 0; k < K_TAPS; ++k) {
    const bool act = (k < 8) || nb10;            // BLOCK_MASK: zero masked taps once
    const int  ci  = fc * K_TAPS + k;
    fr[k] = act ? fr0[ci] : 0.0f;
    fi[k] = act ? fi0[ci] : 0.0f;
    ar[k] = act ? ar0[ci] : 0.0f;
    ai[k] = act ? ai0[ci] : 0.0f;
    hr[k] = 0.0f; hi[k] = 0.0f;
  }
  float mseIn = 1.0f, mseAd = 1.0f, mseMn = 1.0f;

  // ---- prime the async prefetch ring ---------------------------------------
  for (int p = 0; p < DEPTH; ++p)
    issue_prefetch(ldsLane, p, (size_t)fc, micR, micI, refR, refI);

  int c10 = 0;                                   // NLP frame counter (s % 10)
#pragma unroll 1
  for (int s = 0; s < NSTEPS; ++s) {
    wait_async_le28();                           // oldest slot is ready
    const int slot = s & (DEPTH - 1);
    const float4 v = frameBuf[slot * THREADS + tid];   // ds_load_b128
    wait_ds0();                                  // slot consumed -> safe to overwrite
    {                                            // keep 7 newer slots in flight
      int p = s + DEPTH; if (p > NSTEPS - 1) p = NSTEPS - 1;  // tail: clamped dummy
      issue_prefetch(ldsLane, p, (size_t)fc, micR, micI, refR, refI);
    }

    const float mr = v.x, mi = v.y, rr = v.z, ri = v.w;

    // shift-in new reference frame
#pragma unroll
    for (int k = K_TAPS - 1; k > 0; --k) { hr[k] = hr[k - 1]; hi[k] = hi[k - 1]; }
    hr[0] = rr; hi[0] = ri;

    // dual-filter echo estimates (masked taps have zero coefficients)
    float fer = 0.f, fei = 0.f, aer = 0.f, aei = 0.f;
#pragma unroll
    for (int k = 0; k < K_TAPS; ++k) {
      fer += fr[k] * hr[k] + fi[k] * hi[k];
      fei += fr[k] * hi[k] - fi[k] * hr[k];
      aer += ar[k] * hr[k] + ai[k] * hi[k];
      aei += ar[k] * hi[k] - ai[k] * hr[k];
    }
    const float feR = mr - fer, feI = mi - fei;
    const float aeR = mr - aer, aeI = mi - aei;
    const float fP  = feR * feR + feI * feI;
    const float aP  = aeR * aeR + aeI * aeI;
    const bool  sel = (fP >= aP);                // pick better filter per bin
    const float errR = sel ? aeR : feR;
    const float errI = sel ? aeI : feI;
    const float ecR  = sel ? aer : fer;
    const float ecI  = sel ? aei : fei;

    // MSE smoothing, double-talk / recovery flags
    mseIn = 0.97f * mseIn + 0.03f * (mr * mr + mi * mi);
    mseMn = 0.97f * mseMn + 0.03f * fP;
    mseAd = 0.97f * mseAd + 0.03f * aP;
    const bool dt  = (mseAd > 8.0f * mseIn);
    const bool rec = (mseIn > 8.0f * mseMn) && (mseMn < 0.5f * mseAd);

    // IPNLMS update of the adaptive filter (gated by !dt)
    float mag2[K_TAPS];
    float ssum = 1e-10f, tot = 0.0f;
#pragma unroll
    for (int k = 0; k < K_TAPS; ++k) {
      mag2[k] = ar[k] * ar[k] + ai[k] * ai[k];
      ssum += mag2[k];                           // masked taps are exactly zero
      float h2 = hr[k] * hr[k] + hi[k] * hi[k];
      if (k >= 8) h2 = nb10 ? h2 : 0.0f;         // BLOCK_MASK on history energy
      tot += h2;
    }
    const float muN  = 0.5f / (tot + 1.0e-8f + 1e-10f);   // MU/(tot+BETA+1e-10)
    const float invS = 1.0f / ssum;
    const float gate = dt ? 0.0f : 1.0f;
#pragma unroll
    for (int k = 0; k < K_TAPS; ++k) {
      const float kl = inv2nb + 1.5f * mag2[k] * invS;    // (1+a)=1.5
      const float pr = hr[k] * aeR + hi[k] * aeI;
      const float pi = hi[k] * aeR - hr[k] * aeI;
      const float c  = muN * kl;
      float m = gate;
      if (k >= 8) m = nb10 ? m : 0.0f;           // BLOCK_MASK on update
      const float ur = fminf(fmaxf(c * pr, -0.01f), 0.01f) * m;
      const float ui = fminf(fmaxf(c * pi, -0.01f), 0.01f) * m;
      const float nr = ar[k] + ur, ni = ai[k] + ui;
      const float mg = sqrtf(nr * nr + ni * ni + 1e-10f);
      const float sc = (mg > 2.0f) ? (2.0f / mg) : 1.0f;  // clamp |coef| <= 2
      ar[k] = nr * sc; ai[k] = ni * sc;
    }

    // copy adaptive -> main where adaptive won; recovery main -> adaptive
#pragma unroll
    for (int k = 0; k < K_TAPS; ++k) {
      fr[k] = sel ? ar[k] : fr[k];
      fi[k] = sel ? ai[k] : fi[k];
      ar[k] = rec ? fr[k] : ar[k];
      ai[k] = rec ? fi[k] : ai[k];
    }

    // periodic spectral-subtraction NLP (frames 0,10,20,... ; s is uniform)
    float oR = errR, oI = errI;
    if (c10 == 0) {
      const float em = sqrtf(errR * errR + errI * errI + 1e-12f);
      const float xm = sqrtf(ecR * ecR + ecI * ecI);
      const float sp = fmaxf(em - 1.5f * xm, 0.01f * em);
      const float g  = sp / em;
      oR = g * errR; oI = g * errI;
    }
    if (++c10 == 10) c10 = 0;

    // out[b, t, f, {re,im}] -> flat ((b*NT + t)*F + f)*2 ; write-once stream
    if (tid < F_BINS) {
      const size_t ob = ((size_t)((s & 7) * NT + (s >> 3)) * F_BINS + tid) * 2;
      v2f o; o.x = oR; o.y = oI;
      __builtin_nontemporal_store(o, (v2f*)(out + ob));
    }
  }
}

extern "C" void kernel_launch(void* const* d_in, const int* in_sizes, int n_in,
                              void* d_out, int out_size, void* d_ws, size_t ws_size,
                              hipStream_t stream) {
  (void)in_sizes; (void)n_in; (void)d_ws; (void)ws_size; (void)out_size;
  const float* micR = (const float*)d_in[0];
  const float* micI = (const float*)d_in[1];
  const float* refR = (const float*)d_in[2];
  const float* refI = (const float*)d_in[3];
  const float* fr0  = (const float*)d_in[4];
  const float* fi0  = (const float*)d_in[5];
  const float* ar0  = (const float*)d_in[6];
  const float* ai0  = (const float*)d_in[7];
  float* out = (float*)d_out;

  // Single workgroup: the scan carry is one global serial chain; bins are the
  // only parallel axis. 160 threads = 5 wave32s on one WGP.
  aec_ipnlms_kernel<<<dim3(1), dim3(THREADS), 0, stream>>>(
      micR, micI, refR, refI, fr0, fi0, ar0, ai0, out);
}